// TopKRouter_64261300683106
// MI455X (gfx1250) — compile-verified
//
#include <hip/hip_runtime.h>
#include <hip/hip_bf16.h>
#include <math.h>

typedef __attribute__((ext_vector_type(2))) float v2f;
typedef __attribute__((ext_vector_type(8))) float v8f;

namespace {
constexpr int kTokens  = 16384;
constexpr int kDim     = 2048;
constexpr int kExperts = 64;
constexpr int kTopK    = 8;
constexpr int kTokBlk  = 128;   // tokens per block: 8 waves x 16 tokens
constexpr int kStride  = 65;    // padded LDS row -> conflict-free column scans
}

__launch_bounds__(256, 2)
__global__ void topk_router_kernel(const float* __restrict__ input,
                                   const float* __restrict__ weight,
                                   float* __restrict__ out) {
  __shared__ float sLogit[kTokBlk * kStride];
  __shared__ float sProb[kTokBlk * kTopK];
  __shared__ int   sIdx [kTokBlk * kTopK];

  const int tid   = threadIdx.x;
  const int lane  = tid & 31;
  const int wave  = tid >> 5;
  const int m     = lane & 15;   // row (token/expert) within 16-wide tile
  const int khalf = lane >> 4;   // which K-half this lane supplies

  const int row0 = blockIdx.x * kTokBlk + wave * 16;

  // A fragment (input 16x4) and B fragments (weight^T 4x16) share the same
  // lane addressing: lane l covers k + khalf*4 + {0..3}; the (x,y)/(z,w)
  // split feeds two WMMA K-steps with a K-permutation consistent on A and B.
  const float* aPtr  = input  + (size_t)(row0 + m) * kDim + khalf * 4;
  const float* b0Ptr = weight + (size_t)( 0 + m) * kDim + khalf * 4;
  const float* b1Ptr = weight + (size_t)(16 + m) * kDim + khalf * 4;
  const float* b2Ptr = weight + (size_t)(32 + m) * kDim + khalf * 4;
  const float* b3Ptr = weight + (size_t)(48 + m) * kDim + khalf * 4;

  v8f acc0 = {}, acc1 = {}, acc2 = {}, acc3 = {};

  #pragma unroll 2
  for (int k = 0; k < kDim; k += 8) {
    float4 a  = *(const float4*)(aPtr  + k);
    float4 b0 = *(const float4*)(b0Ptr + k);
    float4 b1 = *(const float4*)(b1Ptr + k);
    float4 b2 = *(const float4*)(b2Ptr + k);
    float4 b3 = *(const float4*)(b3Ptr + k);

    v2f aLo = {a.x, a.y};
    v2f aHi = {a.z, a.w};

    v2f t0 = {b0.x, b0.y}, t1 = {b0.z, b0.w};
    acc0 = __builtin_amdgcn_wmma_f32_16x16x4_f32(false, aLo, false, t0, (short)0, acc0, false, false);
    acc0 = __builtin_amdgcn_wmma_f32_16x16x4_f32(false, aHi, false, t1, (short)0, acc0, false, false);

    v2f u0 = {b1.x, b1.y}, u1 = {b1.z, b1.w};
    acc1 = __builtin_amdgcn_wmma_f32_16x16x4_f32(false, aLo, false, u0, (short)0, acc1, false, false);
    acc1 = __builtin_amdgcn_wmma_f32_16x16x4_f32(false, aHi, false, u1, (short)0, acc1, false, false);

    v2f w0 = {b2.x, b2.y}, w1 = {b2.z, b2.w};
    acc2 = __builtin_amdgcn_wmma_f32_16x16x4_f32(false, aLo, false, w0, (short)0, acc2, false, false);
    acc2 = __builtin_amdgcn_wmma_f32_16x16x4_f32(false, aHi, false, w1, (short)0, acc2, false, false);

    v2f x0 = {b3.x, b3.y}, x1 = {b3.z, b3.w};
    acc3 = __builtin_amdgcn_wmma_f32_16x16x4_f32(false, aLo, false, x0, (short)0, acc3, false, false);
    acc3 = __builtin_amdgcn_wmma_f32_16x16x4_f32(false, aHi, false, x1, (short)0, acc3, false, false);
  }

  // C/D layout: VGPR r holds token M=r (lanes 0-15) and M=r+8 (lanes 16-31),
  // expert N = lane & 15 within the tile.
  #pragma unroll
  for (int r = 0; r < 8; ++r) {
    const int tk = wave * 16 + khalf * 8 + r;
    float* dst = &sLogit[tk * kStride + m];
    dst[ 0] = acc0[r];
    dst[16] = acc1[r];
    dst[32] = acc2[r];
    dst[48] = acc3[r];
  }
  __syncthreads();

  // Phase 2: per-token top-8 (lowest-index tie-break, matching jax top_k),
  // then softmax over the selected logits.
  if (tid < kTokBlk) {
    const float* row = &sLogit[tid * kStride];
    unsigned long long chosen = 0ull;
    float v[kTopK];
    int   id[kTopK];
    #pragma unroll 1
    for (int i = 0; i < kTopK; ++i) {
      float best = -3.402823466e38f;
      int   bi   = 0;
      #pragma unroll 1
      for (int e = 0; e < kExperts; ++e) {
        float x = row[e];
        bool free_slot = (((chosen >> e) & 1ull) == 0ull);
        if (free_slot && x > best) { best = x; bi = e; }
      }
      chosen |= (1ull << bi);
      v[i]  = best;
      id[i] = bi;
    }
    const float mx = v[0];           // selection order is descending
    float p[kTopK];
    float s = 0.f;
    #pragma unroll
    for (int i = 0; i < kTopK; ++i) { p[i] = __expf(v[i] - mx); s += p[i]; }
    const float inv = 1.0f / s;
    #pragma unroll
    for (int i = 0; i < kTopK; ++i) {
      sProb[tid * kTopK + i] = p[i] * inv;
      sIdx [tid * kTopK + i] = id[i];
    }
  }
  __syncthreads();

  // Phase 3: coalesced scatter of dense probs + map rows.
  float* probsOut = out;
  float* mapOut   = out + (size_t)kTokens * kExperts;
  const size_t base = (size_t)blockIdx.x * (kTokBlk * kExperts);
  #pragma unroll 1
  for (int it = 0; it < (kTokBlk * kExperts) / 256; ++it) {
    const int idx = it * 256 + tid;
    const int tl  = idx >> 6;
    const int e   = idx & 63;
    float pr = 0.f, mp = 0.f;
    #pragma unroll
    for (int i = 0; i < kTopK; ++i) {
      if (sIdx[tl * kTopK + i] == e) { pr = sProb[tl * kTopK + i]; mp = 1.0f; }
    }
    probsOut[base + idx] = pr;
    mapOut  [base + idx] = mp;
  }
}

extern "C" void kernel_launch(void* const* d_in, const int* in_sizes, int n_in,
                              void* d_out, int out_size, void* d_ws, size_t ws_size,
                              hipStream_t stream) {
  (void)in_sizes; (void)n_in; (void)out_size; (void)d_ws; (void)ws_size;
  const float* input  = (const float*)d_in[0];
  const float* weight = (const float*)d_in[1];
  float* out = (float*)d_out;
  dim3 grid(kTokens / kTokBlk);   // 128 blocks
  dim3 block(256);                // 8 wave32s
  topk_router_kernel<<<grid, block, 0, stream>>>(input, weight, out);
}